// BlockLocalAttention_36206574306110
// MI455X (gfx1250) — compile-verified
//
#include <hip/hip_runtime.h>
#include <math.h>

// ---------------------------------------------------------------------------
// Block-local attention for MI455X (gfx1250, wave32, WMMA).
// - f32 WMMA (V_WMMA_F32_16X16X4_F32): the kernel is HBM-write bound (~369MB
//   of outputs vs ~2.2 GFLOP of matmul), so full-precision matrix ops cost
//   nothing and match the f32 reference exactly.
// - Non-temporal loads/stores: every input tile is read by exactly one
//   workgroup and every output byte is written once and never re-read, and
//   attn_weights alone (256MB) exceeds the 192MB L2 — stream everything
//   (TH=NT) to avoid rinsing dirty RT lines through L2.
// ---------------------------------------------------------------------------

typedef __attribute__((ext_vector_type(2))) float v2f;
typedef __attribute__((ext_vector_type(4))) float f4;
typedef __attribute__((ext_vector_type(8))) float v8f;

#define S_LEN    4096
#define D_MODEL  512
#define BLK      64
#define CHUNK    64
#define PAD      68      // row stride in floats: 68%4==0 (float4 LDS ok),
                         // bank stride 4 -> conflict-free column reads
#define NTHREADS 128
#define ATTN_SCALE 0.04419417382415922f   // 1/sqrt(512)

__global__ __launch_bounds__(NTHREADS)
void block_attn_kernel(const float* __restrict__ q,
                       const float* __restrict__ k,
                       const float* __restrict__ v,
                       float* __restrict__ out)
{
    __shared__ float Qs[BLK][PAD];
    __shared__ float Ks[BLK][PAD];   // reused as Vs in phase 2
    __shared__ float Ss[BLK][PAD];   // scores -> probabilities

    const int blk  = blockIdx.x;     // 0..63 : which 64-token block
    const int b    = blockIdx.y;     // 0..3  : batch
    const int tid  = threadIdx.x;
    const int wave = tid >> 5;       // 4 waves; wave w owns query rows [16w,16w+16)
    const int lane = tid & 31;
    const int lh   = lane >> 4;      // half-wave select
    const int ll   = lane & 15;
    const int q0   = blk * BLK;      // block-diagonal: key base == query base

    float* d_o    = out;
    float* d_attn = out + (long long)4 * S_LEN * D_MODEL;
    float* d_mask = d_attn + (long long)4 * S_LEN * S_LEN;

    const long long tile_base = ((long long)b * S_LEN + q0) * D_MODEL;

    // ---------------- Phase 1: S = (Q K^T) * scale --------------------------
    v8f sacc[4];
    #pragma unroll
    for (int t = 0; t < 4; ++t) sacc[t] = (v8f){0.f,0.f,0.f,0.f,0.f,0.f,0.f,0.f};

    for (int d0 = 0; d0 < D_MODEL; d0 += CHUNK) {
        // Coalesced float4 staging of the 64x64 Q and K chunks (one-touch -> NT).
        #pragma unroll
        for (int it = 0; it < 8; ++it) {
            int i4 = tid + it * NTHREADS;        // 0..1023 float4 slots
            int r  = i4 >> 4;
            int c  = (i4 & 15) << 2;
            *(f4*)&Qs[r][c] = __builtin_nontemporal_load(
                (const f4*)&q[tile_base + (long long)r * D_MODEL + d0 + c]);
            *(f4*)&Ks[r][c] = __builtin_nontemporal_load(
                (const f4*)&k[tile_base + (long long)r * D_MODEL + d0 + c]);
        }
        __syncthreads();

        #pragma unroll 4
        for (int k0 = 0; k0 < CHUNK; k0 += 4) {
            // A fragment: A[m][kk], m = ll, kk = 2*lh + j
            v2f a;
            a.x = Qs[wave * 16 + ll][k0 + 2 * lh + 0];
            a.y = Qs[wave * 16 + ll][k0 + 2 * lh + 1];
            #pragma unroll
            for (int t = 0; t < 4; ++t) {
                // B fragment: B[kk][n] = K[key n][kk], n = t*16+ll
                v2f bm;
                bm.x = Ks[t * 16 + ll][k0 + 2 * lh + 0];
                bm.y = Ks[t * 16 + ll][k0 + 2 * lh + 1];
                sacc[t] = __builtin_amdgcn_wmma_f32_16x16x4_f32(
                    false, a, false, bm, (short)0, sacc[t], false, false);
            }
        }
        __syncthreads();
    }

    // Scaled scores -> LDS (C layout: M = r + 8*lh, N = ll)
    #pragma unroll
    for (int t = 0; t < 4; ++t)
        #pragma unroll
        for (int r = 0; r < 8; ++r)
            Ss[wave * 16 + r + 8 * lh][t * 16 + ll] = sacc[t][r] * ATTN_SCALE;
    __syncthreads();

    // ---------------- Row softmax over the 64 in-block keys -----------------
    if (tid < BLK) {
        float mx = -1e30f;
        for (int j = 0; j < BLK; ++j) mx = fmaxf(mx, Ss[tid][j]);
        float sum = 0.f;
        for (int j = 0; j < BLK; ++j) {
            float e = __expf(Ss[tid][j] - mx);
            Ss[tid][j] = e;
            sum += e;
        }
        float inv = 1.0f / sum;
        for (int j = 0; j < BLK; ++j) Ss[tid][j] *= inv;
    }
    __syncthreads();

    // ------- Dense attn_weights rows (+ mask rows, batch 0 only) ------------
    // Pure streaming writes (never re-read, >L2 capacity): non-temporal.
    {
        const long long rowBase = ((long long)b * S_LEN + q0) * S_LEN;
        const f4 zero = {0.f, 0.f, 0.f, 0.f};
        const f4 ones = {1.f, 1.f, 1.f, 1.f};
        for (int idx = tid; idx < BLK * (S_LEN / 4); idx += NTHREADS) {
            int r = idx >> 10;             // row within block
            int c = (idx & 1023) << 2;     // column (float4 aligned)
            bool inblk = (c >= q0) && (c < q0 + BLK);
            f4 val = inblk ? *(const f4*)&Ss[r][c - q0] : zero;
            __builtin_nontemporal_store(
                val, (f4*)&d_attn[rowBase + (long long)r * S_LEN + c]);
            if (b == 0)
                __builtin_nontemporal_store(
                    inblk ? ones : zero,
                    (f4*)&d_mask[(long long)(q0 + r) * S_LEN + c]);
        }
    }

    // ---------------- Phase 2: O = P @ V ------------------------------------
    for (int d0 = 0; d0 < D_MODEL; d0 += CHUNK) {
        #pragma unroll
        for (int it = 0; it < 8; ++it) {
            int i4 = tid + it * NTHREADS;
            int r  = i4 >> 4;
            int c  = (i4 & 15) << 2;
            *(f4*)&Ks[r][c] = __builtin_nontemporal_load(
                (const f4*)&v[tile_base + (long long)r * D_MODEL + d0 + c]);
        }
        __syncthreads();

        v8f oacc[4];
        #pragma unroll
        for (int t = 0; t < 4; ++t) oacc[t] = (v8f){0.f,0.f,0.f,0.f,0.f,0.f,0.f,0.f};

        #pragma unroll 4
        for (int k0 = 0; k0 < BLK; k0 += 4) {
            // A fragment: P[m][kk]
            v2f a;
            a.x = Ss[wave * 16 + ll][k0 + 2 * lh + 0];
            a.y = Ss[wave * 16 + ll][k0 + 2 * lh + 1];
            #pragma unroll
            for (int t = 0; t < 4; ++t) {
                // B fragment: V[key kk][d0 + t*16 + n]
                v2f bm;
                bm.x = Ks[k0 + 2 * lh + 0][t * 16 + ll];
                bm.y = Ks[k0 + 2 * lh + 1][t * 16 + ll];
                oacc[t] = __builtin_amdgcn_wmma_f32_16x16x4_f32(
                    false, a, false, bm, (short)0, oacc[t], false, false);
            }
        }

        // Coalesced output stores: 16 lanes -> 16 consecutive columns (NT).
        #pragma unroll
        for (int t = 0; t < 4; ++t)
            #pragma unroll
            for (int r = 0; r < 8; ++r)
                __builtin_nontemporal_store(
                    oacc[t][r],
                    &d_o[((long long)b * S_LEN + q0 + wave * 16 + r + 8 * lh) * D_MODEL
                         + d0 + t * 16 + ll]);
        __syncthreads();
    }
}

extern "C" void kernel_launch(void* const* d_in, const int* in_sizes, int n_in,
                              void* d_out, int out_size, void* d_ws, size_t ws_size,
                              hipStream_t stream) {
    (void)in_sizes; (void)n_in; (void)out_size; (void)d_ws; (void)ws_size;
    const float* q = (const float*)d_in[0];
    const float* k = (const float*)d_in[1];
    const float* v = (const float*)d_in[2];
    float* out = (float*)d_out;

    dim3 grid(S_LEN / BLK, 4);   // 64 blocks x 4 batches = 256 workgroups
    block_attn_kernel<<<grid, NTHREADS, 0, stream>>>(q, k, v, out);
}